// SA_1400159338872
// MI455X (gfx1250) — compile-verified
//
#include <hip/hip_runtime.h>
#include <stdint.h>

typedef __attribute__((ext_vector_type(2))) float v2f;
typedef __attribute__((ext_vector_type(8))) float v8f;

#define MV      4096          // voxels
#define NP      32768         // points
#define NS      16            // samples per voxel
#define R2      0.01f         // radius^2
#define R2BAND  (0.01f + 1e-5f)  // WMMA prune band (covers expansion rounding)
#define EPSV    1e-5f
#define LROWS   (MV * NS)     // 65536 MLP rows
#define NBLK    (MV / 8)      // 512 blocks for the layer kernels

// ---------------------------------------------------------------------------
// Kernel 1: ball query. WMMA f32 16x16x4 computes the *expanded* squared
// distance as a candidate filter (band-widened); every candidate is then
// re-verified with the exact difference form (matches the reference's
// sum((c-p)^2) formulation) before being appended. Block = 4 waves; each wave
// scans an index-ordered quarter of the point tiles; ordered lists merged in
// LDS. Point-tile loads are software-pipelined one iteration ahead.
// ---------------------------------------------------------------------------
__global__ __launch_bounds__(128) void sa_ballquery(
    const float* __restrict__ vc, const float* __restrict__ pts,
    int* __restrict__ idx_out)
{
    __shared__ int s_idx[4][16][16];
    __shared__ int s_cnt[4][16];

    const int lane  = threadIdx.x & 31;
    const int wave  = threadIdx.x >> 5;
    const int hi    = lane >> 4;      // which K-half this lane holds
    const int lr    = lane & 15;      // row / col within tile
    const int tileM = blockIdx.x * 16;

    // voxel-center coords for row lr (lanes 0-15 and 16-31 mirror)
    const int m = tileM + lr;
    const float cx = vc[m * 3 + 0];
    const float cy = vc[m * 3 + 1];
    const float cz = vc[m * 3 + 2];
    const float cn = cx * cx + cy * cy + cz * cz;

    // A fragment (16x4 f32): VGPR0 = K(0|2), VGPR1 = K(1|3)
    v2f a;
    a.x = hi ? cz : cx;
    a.y = hi ? 1.0f : cy;

    // C fragment: row-constant |c|^2 ; VGPR v holds rows v (lo) / v+8 (hi)
    v8f cacc;
#pragma unroll
    for (int v = 0; v < 8; ++v)
        cacc[v] = __shfl(cn, v + 8 * hi, 32);

    int cnt = 0;                       // neighbor count (valid in lanes 0-15)
    const int NT    = NP / 16 / 4;     // 512 tiles per wave segment
    const int tile0 = wave * NT;

    float4 pnext = ((const float4*)pts)[tile0 * 16 + lr];
    for (int t = 0; t < NT; ++t) {
        const int nbase = (tile0 + t) * 16;
        const float4 p = pnext;
        if (t + 1 < NT)                 // prefetch next tile past the WMMA
            pnext = ((const float4*)pts)[nbase + 16 + lr];

        const float pn2 = p.x * p.x + p.y * p.y + p.z * p.z;
        v2f b;
        b.x = hi ? (-2.0f * p.z) : (-2.0f * p.x);
        b.y = hi ? pn2 : (-2.0f * p.y);

        v8f d = __builtin_amdgcn_wmma_f32_16x16x4_f32(
            false, a, false, b, (short)0, cacc, false, false);

        // Row r's 16 candidate bits live in ballot(v=r&7), half r>>3.
        uint32_t mymask = 0;
#pragma unroll
        for (int v = 0; v < 8; ++v) {
            uint32_t bal = (uint32_t)__ballot(d[v] < R2BAND);
            if ((lr & 7) == v)
                mymask = (lr >> 3) ? (bal >> 16) : (bal & 0xffffu);
        }

        if (!hi) {
            while (mymask && cnt < NS) {
                const int bit = __ffs(mymask) - 1;
                mymask &= mymask - 1;
                const int pi = nbase + bit;
                // exact difference-form check (reference formulation)
                const float4 q = ((const float4*)pts)[pi];
                const float dx = cx - q.x, dy = cy - q.y, dz = cz - q.z;
                if (dx * dx + dy * dy + dz * dz < R2) {
                    s_idx[wave][lr][cnt] = pi;
                    ++cnt;
                }
            }
        }
        // uniform early exit once all 16 rows of this segment are full
        const uint32_t doneb = (uint32_t)__ballot(cnt >= NS);
        if ((doneb & 0xffffu) == 0xffffu) break;
    }
    if (!hi) s_cnt[wave][lr] = cnt;
    __syncthreads();

    // merge the 4 ordered segment lists; pad with first index; empty -> -1
    if (threadIdx.x < 16) {
        const int r = threadIdx.x;
        int merged[NS];
        int total = 0;
        for (int seg = 0; seg < 4; ++seg) {
            const int c = s_cnt[seg][r];
            for (int j = 0; j < c && total < NS; ++j)
                merged[total++] = s_idx[seg][r][j];
        }
        if (total == 0) {
            for (int s = 0; s < NS; ++s) merged[s] = -1;   // empty sentinel
        } else {
            for (int s = total; s < NS; ++s) merged[s] = merged[0];
        }
        for (int s = 0; s < NS; ++s)
            idx_out[(tileM + r) * NS + s] = merged[s];
    }
}

// ---------------------------------------------------------------------------
// Deterministic fixed-order reduction of per-block BN partials.
// ---------------------------------------------------------------------------
__global__ void sa_reduce2(const float* __restrict__ psum,
                           const float* __restrict__ psq,
                           float* __restrict__ osum, float* __restrict__ osq,
                           int nb, int C)
{
    const int tid = threadIdx.x;
    if (tid < C) {
        float s = 0.f;
        for (int b = 0; b < nb; ++b) s += psum[b * C + tid];
        osum[tid] = s;
    } else if (tid < 2 * C) {
        const int c = tid - C;
        float s = 0.f;
        for (int b = 0; b < nb; ++b) s += psq[b * C + c];
        osq[c] = s;
    }
}

// ---------------------------------------------------------------------------
// Kernel 2: gather + layer0 GEMM (K=4) + deterministic BN partials.
// One wave per voxel (16 rows). Block = 256 threads = 8 waves.
// ---------------------------------------------------------------------------
__global__ __launch_bounds__(256) void sa_layer0(
    const float* __restrict__ vc, const float* __restrict__ pts,
    const int* __restrict__ idx, const float* __restrict__ w0,
    float* __restrict__ h0, float* __restrict__ psum, float* __restrict__ psq)
{
    __shared__ float wsum[8][16], wsq[8][16];

    const int lane = threadIdx.x & 31;
    const int wave = threadIdx.x >> 5;
    const int hi = lane >> 4, lr = lane & 15;
    const int t = blockIdx.x * 8 + wave;        // voxel / row-tile id

    const int i = idx[t * NS + lr];
    float4 p = make_float4(0.f, 0.f, 0.f, 0.f);
    if (i >= 0) p = ((const float4*)pts)[i];
    const float gx = vc[t * 3 + 0], gy = vc[t * 3 + 1], gz = vc[t * 3 + 2];

    v2f a;
    if (i >= 0) {
        a.x = hi ? (p.z - gz) : (p.x - gx);
        a.y = hi ? p.w : (p.y - gy);
    } else {
        a.x = 0.f; a.y = 0.f;
    }
    v2f b;                                      // B[k][o] = w0[o*4+k], o = lr
    b.x = w0[lr * 4 + (hi ? 2 : 0)];
    b.y = w0[lr * 4 + (hi ? 3 : 1)];

    v8f acc = {};
    acc = __builtin_amdgcn_wmma_f32_16x16x4_f32(
        false, a, false, b, (short)0, acc, false, false);

    float s1 = 0.f, s2 = 0.f;
#pragma unroll
    for (int v = 0; v < 8; ++v) {
        const float val = acc[v];
        h0[(t * 16 + v + 8 * hi) * 16 + lr] = val;
        s1 += val; s2 += val * val;
    }
    s1 += __shfl_xor(s1, 16, 32);
    s2 += __shfl_xor(s2, 16, 32);
    if (!hi) { wsum[wave][lr] = s1; wsq[wave][lr] = s2; }
    __syncthreads();
    if (threadIdx.x < 16) {
        float a1 = 0.f, a2 = 0.f;
        for (int w = 0; w < 8; ++w) { a1 += wsum[w][threadIdx.x]; a2 += wsq[w][threadIdx.x]; }
        psum[blockIdx.x * 16 + threadIdx.x] = a1;
        psq[blockIdx.x * 16 + threadIdx.x] = a2;
    }
}

// ---------------------------------------------------------------------------
// Kernel 3: hidden layer 16->16: BN(prev)+ReLU fused into A, K=16 as 4
// chained 16x16x4 f32 WMMAs; deterministic BN partials for the next layer.
// ---------------------------------------------------------------------------
__global__ __launch_bounds__(256) void sa_layer16(
    const float* __restrict__ hin, const float* __restrict__ w,
    const float* __restrict__ gma, const float* __restrict__ bta,
    const float* __restrict__ isum, const float* __restrict__ isq,
    float* __restrict__ hout, float* __restrict__ psum, float* __restrict__ psq)
{
    __shared__ float sc[16], sh[16], wsum[8][16], wsq[8][16];
    if (threadIdx.x < 16) {
        const int k = threadIdx.x;
        const float mu  = isum[k] * (1.0f / LROWS);
        const float var = isq[k] * (1.0f / LROWS) - mu * mu;
        const float s   = rsqrtf(var + EPSV) * gma[k];
        sc[k] = s; sh[k] = bta[k] - mu * s;
    }
    __syncthreads();

    const int lane = threadIdx.x & 31;
    const int wave = threadIdx.x >> 5;
    const int hi = lane >> 4, lr = lane & 15;
    const int t = blockIdx.x * 8 + wave;
    const int row = t * 16 + lr;

    v8f acc = {};
#pragma unroll
    for (int kc = 0; kc < 4; ++kc) {
        const int k0 = kc * 4 + (hi ? 2 : 0);
        const int k1 = kc * 4 + (hi ? 3 : 1);
        const float x0 = fmaxf(hin[row * 16 + k0] * sc[k0] + sh[k0], 0.f);
        const float x1 = fmaxf(hin[row * 16 + k1] * sc[k1] + sh[k1], 0.f);
        v2f a; a.x = x0; a.y = x1;
        v2f b; b.x = w[lr * 16 + k0]; b.y = w[lr * 16 + k1];
        acc = __builtin_amdgcn_wmma_f32_16x16x4_f32(
            false, a, false, b, (short)0, acc, false, false);
    }

    float s1 = 0.f, s2 = 0.f;
#pragma unroll
    for (int v = 0; v < 8; ++v) {
        const float val = acc[v];
        hout[(t * 16 + v + 8 * hi) * 16 + lr] = val;
        s1 += val; s2 += val * val;
    }
    s1 += __shfl_xor(s1, 16, 32);
    s2 += __shfl_xor(s2, 16, 32);
    if (!hi) { wsum[wave][lr] = s1; wsq[wave][lr] = s2; }
    __syncthreads();
    if (threadIdx.x < 16) {
        float a1 = 0.f, a2 = 0.f;
        for (int w8 = 0; w8 < 8; ++w8) { a1 += wsum[w8][threadIdx.x]; a2 += wsq[w8][threadIdx.x]; }
        psum[blockIdx.x * 16 + threadIdx.x] = a1;
        psq[blockIdx.x * 16 + threadIdx.x] = a2;
    }
}

// ---------------------------------------------------------------------------
// Kernel 4: layer 16->32: two output-column tiles sharing A fragments.
// ---------------------------------------------------------------------------
__global__ __launch_bounds__(256) void sa_layer32(
    const float* __restrict__ hin, const float* __restrict__ w,
    const float* __restrict__ gma, const float* __restrict__ bta,
    const float* __restrict__ isum, const float* __restrict__ isq,
    float* __restrict__ hout, float* __restrict__ psum, float* __restrict__ psq)
{
    __shared__ float sc[16], sh[16], wsum[8][32], wsq[8][32];
    if (threadIdx.x < 16) {
        const int k = threadIdx.x;
        const float mu  = isum[k] * (1.0f / LROWS);
        const float var = isq[k] * (1.0f / LROWS) - mu * mu;
        const float s   = rsqrtf(var + EPSV) * gma[k];
        sc[k] = s; sh[k] = bta[k] - mu * s;
    }
    __syncthreads();

    const int lane = threadIdx.x & 31;
    const int wave = threadIdx.x >> 5;
    const int hi = lane >> 4, lr = lane & 15;
    const int t = blockIdx.x * 8 + wave;
    const int row = t * 16 + lr;

    v8f acc0 = {};
    v8f acc1 = {};
#pragma unroll
    for (int kc = 0; kc < 4; ++kc) {
        const int k0 = kc * 4 + (hi ? 2 : 0);
        const int k1 = kc * 4 + (hi ? 3 : 1);
        const float x0 = fmaxf(hin[row * 16 + k0] * sc[k0] + sh[k0], 0.f);
        const float x1 = fmaxf(hin[row * 16 + k1] * sc[k1] + sh[k1], 0.f);
        v2f a; a.x = x0; a.y = x1;
        v2f b0; b0.x = w[lr * 16 + k0];        b0.y = w[lr * 16 + k1];        // o = lr
        v2f b1; b1.x = w[(lr + 16) * 16 + k0]; b1.y = w[(lr + 16) * 16 + k1]; // o = lr+16
        acc0 = __builtin_amdgcn_wmma_f32_16x16x4_f32(
            false, a, false, b0, (short)0, acc0, false, false);
        acc1 = __builtin_amdgcn_wmma_f32_16x16x4_f32(
            false, a, false, b1, (short)0, acc1, false, false);
    }

    float s1a = 0.f, s2a = 0.f, s1b = 0.f, s2b = 0.f;
#pragma unroll
    for (int v = 0; v < 8; ++v) {
        const int r = t * 16 + v + 8 * hi;
        const float va = acc0[v];
        const float vb = acc1[v];
        hout[r * 32 + lr]      = va;
        hout[r * 32 + lr + 16] = vb;
        s1a += va; s2a += va * va;
        s1b += vb; s2b += vb * vb;
    }
    s1a += __shfl_xor(s1a, 16, 32); s2a += __shfl_xor(s2a, 16, 32);
    s1b += __shfl_xor(s1b, 16, 32); s2b += __shfl_xor(s2b, 16, 32);
    if (!hi) {
        wsum[wave][lr] = s1a;      wsq[wave][lr] = s2a;
        wsum[wave][lr + 16] = s1b; wsq[wave][lr + 16] = s2b;
    }
    __syncthreads();
    if (threadIdx.x < 32) {
        float a1 = 0.f, a2 = 0.f;
        for (int w8 = 0; w8 < 8; ++w8) { a1 += wsum[w8][threadIdx.x]; a2 += wsq[w8][threadIdx.x]; }
        psum[blockIdx.x * 32 + threadIdx.x] = a1;
        psq[blockIdx.x * 32 + threadIdx.x] = a2;
    }
}

// ---------------------------------------------------------------------------
// Kernel 5: final BN+ReLU + max over the 16 samples -> out[4096][32].
// ---------------------------------------------------------------------------
__global__ void sa_final(
    const float* __restrict__ h2, const float* __restrict__ gma,
    const float* __restrict__ bta, const float* __restrict__ isum,
    const float* __restrict__ isq, float* __restrict__ out)
{
    const int tid = blockIdx.x * blockDim.x + threadIdx.x;
    if (tid >= MV * 32) return;
    const int ch = tid & 31, m = tid >> 5;
    const float mu  = isum[ch] * (1.0f / LROWS);
    const float var = isq[ch] * (1.0f / LROWS) - mu * mu;
    const float s   = rsqrtf(var + EPSV) * gma[ch];
    const float sh  = bta[ch] - mu * s;
    float mx = 0.0f;                       // post-ReLU values are >= 0
    for (int k = 0; k < NS; ++k) {
        const float v = fmaxf(h2[(m * NS + k) * 32 + ch] * s + sh, 0.f);
        mx = fmaxf(mx, v);
    }
    out[m * 32 + ch] = mx;
}

// ---------------------------------------------------------------------------
extern "C" void kernel_launch(void* const* d_in, const int* in_sizes, int n_in,
                              void* d_out, int out_size, void* d_ws, size_t ws_size,
                              hipStream_t stream)
{
    const float* vc  = (const float*)d_in[0];   // voxel_center (M,3)
    // d_in[1] = voxel_coords (unused)
    const float* pts = (const float*)d_in[2];   // pts (N,4)
    const float* w0  = (const float*)d_in[3];
    const float* g0  = (const float*)d_in[4];
    const float* b0  = (const float*)d_in[5];
    const float* w1  = (const float*)d_in[6];
    const float* g1  = (const float*)d_in[7];
    const float* b1  = (const float*)d_in[8];
    const float* w2  = (const float*)d_in[9];
    const float* g2  = (const float*)d_in[10];
    const float* b2  = (const float*)d_in[11];
    float* out = (float*)d_out;

    // workspace layout (every region fully rewritten each call)
    char* ws = (char*)d_ws;
    int*   idx = (int*)ws;                                   // 65536 ints
    float* h0  = (float*)(ws + (size_t)LROWS * sizeof(int)); // 65536*16 f32
    float* h1  = h0 + (size_t)LROWS * 16;
    float* h2  = h1 + (size_t)LROWS * 16;                    // 65536*32 f32
    float* st  = h2 + (size_t)LROWS * 32;                    // 128 stat floats
    float* s0sum = st;       float* s0sq = st + 16;
    float* s1sum = st + 32;  float* s1sq = st + 48;
    float* s2sum = st + 64;  float* s2sq = st + 96;
    float* p0sum = st + 128;            float* p0sq = p0sum + NBLK * 16;
    float* p1sum = p0sq + NBLK * 16;    float* p1sq = p1sum + NBLK * 16;
    float* p2sum = p1sq + NBLK * 16;    float* p2sq = p2sum + NBLK * 32;

    sa_ballquery<<<MV / 16, 128, 0, stream>>>(vc, pts, idx);
    sa_layer0<<<NBLK, 256, 0, stream>>>(vc, pts, idx, w0, h0, p0sum, p0sq);
    sa_reduce2<<<1, 64, 0, stream>>>(p0sum, p0sq, s0sum, s0sq, NBLK, 16);
    sa_layer16<<<NBLK, 256, 0, stream>>>(h0, w1, g0, b0, s0sum, s0sq, h1, p1sum, p1sq);
    sa_reduce2<<<1, 64, 0, stream>>>(p1sum, p1sq, s1sum, s1sq, NBLK, 16);
    sa_layer32<<<NBLK, 256, 0, stream>>>(h1, w2, g1, b1, s1sum, s1sq, h2, p2sum, p2sq);
    sa_reduce2<<<1, 64, 0, stream>>>(p2sum, p2sq, s2sum, s2sq, NBLK, 32);
    sa_final<<<(MV * 32 + 255) / 256, 256, 0, stream>>>(h2, g2, b2, s2sum, s2sq, out);
}